// GAEReturns_32375463477693
// MI455X (gfx1250) — compile-verified
//
#include <hip/hip_runtime.h>

typedef float v2f __attribute__((ext_vector_type(2)));
typedef float v8f __attribute__((ext_vector_type(8)));

#define GAMMA_F 0.98f
#define LAM_F   0.95f

// ---------------------------------------------------------------------------
// Kernel A: v_all[t] = relu(states[t,:]@W1 + b1) @ W2 + b2  for t in [0, nRows)
// One wave per 16-row tile, fp32 WMMA 16x16x4, K=64 (16 k-steps), N=64 (4 tiles).
// W1 is staged in LDS pre-swizzled into B-fragment layout: each (kt,n,lane)
// fragment is one aligned float2 -> single ds_load_b64, no repacking movs.
// ---------------------------------------------------------------------------
__global__ __launch_bounds__(256) void gae_mlp_value_kernel(
    const float* __restrict__ states, const float* __restrict__ W1,
    const float* __restrict__ b1, const float* __restrict__ W2,
    const float* __restrict__ b2, float* __restrict__ v_all,
    int nRows, int numTiles)
{
    __shared__ v2f   W1swz[16 * 4 * 32];    // [kt][n][lane] B fragments, 16 KB
    __shared__ float b1s[64];
    __shared__ float W2s[64];
    __shared__ float Atile[8][16 * 68];     // per-wave A tile, stride 68: conflict-free frag reads

    const int tid = threadIdx.x;

    // Stage W1 swizzled into B-fragment layout:
    //   frag(kt,n,lane) = { W1[kt*4 + 0 + 2*(lane/16)][n*16 + lane%16],
    //                       W1[kt*4 + 1 + 2*(lane/16)][n*16 + lane%16] }
    for (int i = tid; i < 16 * 4 * 32; i += 256) {
        const int lane_i = i & 31;
        const int n_i    = (i >> 5) & 3;
        const int kt_i   = i >> 7;
        const int half_i = lane_i >> 4;
        const int c_i    = lane_i & 15;
        v2f f;
        f.x = W1[(kt_i * 4 + 0 + 2 * half_i) * 64 + n_i * 16 + c_i];
        f.y = W1[(kt_i * 4 + 1 + 2 * half_i) * 64 + n_i * 16 + c_i];
        W1swz[i] = f;
    }
    if (tid < 64) { b1s[tid] = b1[tid]; W2s[tid] = W2[tid]; }
    __syncthreads();

    const float bias2 = b2[0];
    const int wave = tid >> 5;
    const int lane = tid & 31;
    const int half = lane >> 4;     // 0 or 1
    const int cIdx = lane & 15;     // N within n-tile / row within A tile

    const int tile = blockIdx.x * 8 + wave;
    if (tile >= numTiles) return;

    float* At = &Atile[wave][0];
    const v2f* Bfrag = &W1swz[lane];        // index by (kt*4 + n)*32

    // Stage 16x64 fp32 A tile, coalesced float4 loads (2 rows / instruction)
    #pragma unroll
    for (int j = 0; j < 8; ++j) {
        const int idx = j * 32 + lane;
        const int r  = idx >> 4;            // 0..15
        const int c4 = idx & 15;            // float4 column
        long grow = (long)tile * 16 + r;
        if (grow > (long)nRows - 1) grow = nRows - 1;   // clamp tail (stores guarded)
        const float4 v = *(const float4*)(states + grow * 64 + c4 * 4);
        *(float4*)(At + r * 68 + c4 * 4) = v;
    }
    // Same-wave LDS ops are in-order; compiler inserts DScnt waits before reads.

    v8f acc[4];
    #pragma unroll
    for (int n = 0; n < 4; ++n)
        #pragma unroll
        for (int g = 0; g < 8; ++g) acc[n][g] = 0.0f;

    // K loop: 16 steps of K=4; A frag reused across the 4 N-tiles
    for (int kt = 0; kt < 16; ++kt) {
        const int k0 = kt * 4 + 2 * half;
        // word addr cIdx*68 + k0 is even -> aligned 8-byte LDS load
        const v2f a = *(const v2f*)(At + cIdx * 68 + k0);
        #pragma unroll
        for (int n = 0; n < 4; ++n) {
            const v2f bf = Bfrag[(kt * 4 + n) * 32];   // one ds_load_b64
            acc[n] = __builtin_amdgcn_wmma_f32_16x16x4_f32(
                false, a, false, bf, (short)0, acc[n], false, false);
        }
    }

    // Layer 2: v[M] = sum_N relu(h[M,N] + b1[N]) * W2[N] + b2
    float vpart[8];
    #pragma unroll
    for (int g = 0; g < 8; ++g) vpart[g] = 0.0f;
    #pragma unroll
    for (int n = 0; n < 4; ++n) {
        const float bias = b1s[n * 16 + cIdx];
        const float w2v  = W2s[n * 16 + cIdx];
        #pragma unroll
        for (int g = 0; g < 8; ++g) {
            float h = acc[n][g] + bias;
            h = h > 0.0f ? h : 0.0f;
            vpart[g] += h * w2v;
        }
    }
    // Reduce across the 16 lanes of each half (xor butterfly stays in-half for off<16)
    #pragma unroll
    for (int g = 0; g < 8; ++g) {
        float s = vpart[g];
        #pragma unroll
        for (int off = 1; off < 16; off <<= 1)
            s += __shfl_xor(s, off, 32);
        vpart[g] = s;
    }
    if (cIdx == 0) {
        #pragma unroll
        for (int g = 0; g < 8; ++g) {
            const long row = (long)tile * 16 + g + 8 * half;   // C/D layout: M = g + 8*half
            if (row < nRows) v_all[row] = vpart[g] + bias2;
        }
    }
}

// ---------------------------------------------------------------------------
// Kernel B: deltas + reverse linear scan.
// Block owns 4096 outputs (threads 0..255, 16 each) + 1024-element right halo
// (threads 256..319). c^1024 ~ 1e-32 -> halo-seeded carry is exact in fp32.
// ---------------------------------------------------------------------------
__global__ __launch_bounds__(320) void gae_scan_kernel(
    const float* __restrict__ v_all, const float* __restrict__ rewards,
    float* __restrict__ out, int T)
{
    __shared__ float Ssum[320];
    __shared__ float Scarry[320];

    const float decay = GAMMA_F * LAM_F;
    float c16 = 1.0f;
    #pragma unroll
    for (int i = 0; i < 16; ++i) c16 *= decay;

    const int  tid  = threadIdx.x;
    const long base = (long)blockIdx.x * 4096;
    const long s    = base + (long)tid * 16;

    float dl[16];
    #pragma unroll
    for (int j = 0; j < 16; ++j) {
        const long t = s + j;
        if (t < T) {
            const float v  = v_all[t];
            const float vp = v_all[t + 1];
            dl[j] = GAMMA_F * vp + rewards[t] - v;
        } else {
            dl[j] = 0.0f;
        }
    }

    // Segment scan with zero carry -> segment value at its first element
    float a0 = 0.0f;
    #pragma unroll
    for (int j = 15; j >= 0; --j) a0 = dl[j] + decay * a0;
    Ssum[tid] = a0;
    __syncthreads();

    // Chain carries right-to-left across the 320 segments (includes halo)
    if (tid == 0) {
        float cur = 0.0f;
        for (int i = 319; i >= 0; --i) {
            Scarry[i] = cur;             // true adv at first element of segment i+1
            cur = Ssum[i] + c16 * cur;   // true adv at first element of segment i
        }
    }
    __syncthreads();

    if (tid < 256) {
        float adv = Scarry[tid];
        #pragma unroll
        for (int j = 15; j >= 0; --j) {
            adv = dl[j] + decay * adv;
            const long t = s + j;
            if (t < T) out[t] = adv;
        }
    }
}

// ---------------------------------------------------------------------------
extern "C" void kernel_launch(void* const* d_in, const int* in_sizes, int n_in,
                              void* d_out, int out_size, void* d_ws, size_t ws_size,
                              hipStream_t stream) {
    const float* states  = (const float*)d_in[0];
    const float* rewards = (const float*)d_in[1];
    const float* W1      = (const float*)d_in[2];
    const float* b1      = (const float*)d_in[3];
    const float* W2      = (const float*)d_in[4];
    const float* b2      = (const float*)d_in[5];
    float* out = (float*)d_out;

    const int T     = in_sizes[1];       // rewards has T elements
    const int nRows = T + 1;             // v_all entries
    float* v_all = (float*)d_ws;         // (T+1) * 4 bytes of scratch

    const int numTiles = (nRows + 15) / 16;
    const int gridA = (numTiles + 7) / 8;            // 8 waves (tiles) per block
    gae_mlp_value_kernel<<<gridA, 256, 0, stream>>>(
        states, W1, b1, W2, b2, v_all, nRows, numTiles);

    const int gridB = (T + 4095) / 4096;
    gae_scan_kernel<<<gridB, 320, 0, stream>>>(v_all, rewards, out, T);
}